// BlockCNN_24592982737353
// MI455X (gfx1250) — compile-verified
//
#include <hip/hip_runtime.h>
#include <hip/hip_bf16.h>

typedef __attribute__((ext_vector_type(2))) float v2f;
typedef __attribute__((ext_vector_type(8))) float v8f;

#define NT 256

// LDS layout: ~263 KB of the 320 KB per WGP. One workgroup per WGP.
// In (conv1 input tile) and H2 (conv2 output) alias: In is dead after conv1,
// separated from H2 writes by a workgroup barrier.
struct __align__(16) SMem {
  float W1[16 * 500];   // conv1 weights, K = tap*4 + c, Cin padded 3->4 (c=3 col = 0)
  float W2[16 * 432];   // conv2 weights, K = tap*16 + ci
  float W3[16 * 432];   // conv3 weights, rows 3..15 zeroed
  float B1[16], B2[16], B3[16];
  int   Off1[500];      // patch base offsets (float-element units) per k; off[k+1]=off[k]+1
  int   Off2[432];
  int   Off3[432];
  float H1[1728 * 16];  // conv1 out, [12^3 voxel][16 ch]
  union {
    float In[4096 * 4]; // wrap-padded input tile, [16^3 voxel][4 ch] (ch 3 = 0)
    float H2[1000 * 16];// conv2 out, [10^3 voxel][16 ch]
  } u;
};

// Two 16x16 (M=Cout x N=voxels) tiles of implicit-GEMM conv, K accumulated in
// steps of 4 via V_WMMA_F32_16X16X4_F32. The A operand (weights) and the k
// offset are shared by both tiles; the two accumulator chains are independent,
// letting the matrix pipe interleave them (hides D->C accumulate latency).
// A layout (16x4 f32): lane m=lane&15; VGPR0 = W[m][kq],  VGPR1 = W[m][kq+1], kq=k0+2*laneHi
// B layout (4x16 f32): lane n=lane&15; VGPR0 = P[kq][n], VGPR1 = P[kq+1][n]
// K-orders are chosen so P[kq][n] and P[kq+1][n] are adjacent floats in LDS ->
// A and B are single 8B-aligned ds_load_b64 per lane per WMMA.
__device__ __forceinline__ void gemm_acc2(const float* sW, int Kpad,
                                          const int* sOff, const float* sSrc,
                                          int nBase0, int nBase1, int kSteps,
                                          int laneHi, int m, v8f& acc0, v8f& acc1)
{
  const float* wRow = sW + m * Kpad + 2 * laneHi;   // 8B-aligned (Kpad even)
  const int*   oP   = sOff + 2 * laneHi;
#pragma unroll 4
  for (int ks = 0; ks < kSteps; ++ks) {
    const int k0 = ks * 4;
    v2f a  = *(const v2f*)(wRow + k0);              // ds_load_b64 (pairs merge to 2addr)
    int o0 = oP[k0];
    v2f b0 = *(const v2f*)(sSrc + nBase0 + o0);     // ds_load_b64 (contiguous pair)
    v2f b1 = *(const v2f*)(sSrc + nBase1 + o0);     // ds_load_b64
    acc0 = __builtin_amdgcn_wmma_f32_16x16x4_f32(
        false, a, false, b0, (short)0, acc0, false, false);
    acc1 = __builtin_amdgcn_wmma_f32_16x16x4_f32(
        false, a, false, b1, (short)0, acc1, false, false);
  }
}

__global__ __launch_bounds__(NT) void fused_cnn(
    const float* __restrict__ x,
    const float* __restrict__ W1, const float* __restrict__ b1,
    const float* __restrict__ W2, const float* __restrict__ b2,
    const float* __restrict__ W3, const float* __restrict__ b3,
    float* __restrict__ out)
{
  __shared__ SMem sm;
  const int tid = threadIdx.x;
  const int blk = blockIdx.x;
  const int b   = blk / 1728;
  const int t   = blk % 1728;
  const int Od  = (t / 144) * 8;
  const int Oh  = ((t / 12) % 12) * 8;
  const int Ow  = (t % 12) * 8;

  // ---- stage weights / bias / gather tables into LDS ----
  for (int i = tid; i < 16 * 500; i += NT) {
    int co = i / 500, kk = i - co * 500;
    int tap = kk >> 2, c = kk & 3;
    float v = 0.0f;
    if (c < 3) v = W1[(co * 3 + c) * 125 + tap];
    sm.W1[i] = v;
  }
  for (int i = tid; i < 16 * 432; i += NT) {
    int co = i / 432, kk = i - co * 432;
    int ci = kk & 15, tap = kk >> 4;
    sm.W2[i] = W2[(co * 16 + ci) * 27 + tap];
    float w3v = 0.0f;
    if (co < 3) w3v = W3[(co * 16 + ci) * 27 + tap];
    sm.W3[i] = w3v;
  }
  if (tid < 16) {
    sm.B1[tid] = b1[tid];
    sm.B2[tid] = b2[tid];
    sm.B3[tid] = (tid < 3) ? b3[tid] : 0.0f;
  }
  for (int i = tid; i < 500; i += NT) {
    int tap = i >> 2, c = i & 3;
    int kd = tap / 25, r = tap - kd * 25, kh = r / 5, kw = r - kh * 5;
    sm.Off1[i] = (kd * 256 + kh * 16 + kw) * 4 + c;
  }
  for (int i = tid; i < 432; i += NT) {
    int ci = i & 15, tap = i >> 4;
    int kd = tap / 9, r = tap - kd * 9, kh = r / 3, kw = r - kh * 3;
    sm.Off2[i] = (kd * 144 + kh * 12 + kw) * 16 + ci;
    sm.Off3[i] = (kd * 100 + kh * 10 + kw) * 16 + ci;
  }
  // ---- wrap-padded input patch: original coords [O-4, O+12) per dim,
  //      stored [vox][4] with channel 3 zeroed ----
  for (int i = tid; i < 4096 * 4; i += NT) {
    int c = i & 3, v = i >> 2;
    int dz = v >> 8, hy = (v >> 4) & 15, wx = v & 15;
    float val = 0.0f;
    if (c < 3) {
      int gd = Od - 4 + dz; gd += (gd < 0) ? 96 : 0; gd -= (gd >= 96) ? 96 : 0;
      int gh = Oh - 4 + hy; gh += (gh < 0) ? 96 : 0; gh -= (gh >= 96) ? 96 : 0;
      int gw = Ow - 4 + wx; gw += (gw < 0) ? 96 : 0; gw -= (gw >= 96) ? 96 : 0;
      val = x[(size_t)(b * 3 + c) * 884736 + (size_t)gd * 9216 + gh * 96 + gw];
    }
    sm.u.In[i] = val;
  }
  __syncthreads();

  const int lane   = tid & 31;
  const int wave   = tid >> 5;
  const int laneHi = lane >> 4;
  const int ln     = lane & 15;

  // ---- conv1: 5^3, C4->16, over 12^3 region (54 tile-pairs, K=500 -> 125 steps) ----
  {
    v8f cinit;
#pragma unroll
    for (int r = 0; r < 8; ++r) cinit[r] = sm.B1[8 * laneHi + r];
    for (int p = wave; p < 54; p += 8) {
      int vox0 = p * 32 + ln;
      int vox1 = vox0 + 16;
      int d0 = vox0 / 144, r0 = vox0 - d0 * 144, h0 = r0 / 12, w0 = r0 - h0 * 12;
      int d1 = vox1 / 144, r1 = vox1 - d1 * 144, h1 = r1 / 12, w1 = r1 - h1 * 12;
      int nB0 = (d0 * 256 + h0 * 16 + w0) * 4;
      int nB1 = (d1 * 256 + h1 * 16 + w1) * 4;
      v8f a0 = cinit, a1 = cinit;
      gemm_acc2(sm.W1, 500, sm.Off1, sm.u.In, nB0, nB1, 125, laneHi, ln, a0, a1);
      float* dst0 = sm.H1 + vox0 * 16 + 8 * laneHi;
      float* dst1 = sm.H1 + vox1 * 16 + 8 * laneHi;
#pragma unroll
      for (int r = 0; r < 8; ++r) { dst0[r] = a0[r]; dst1[r] = a1[r]; }
    }
  }
  __syncthreads();   // In reads done; safe to overwrite aliased H2

  // ---- conv2: 3^3, 16->16, over 10^3 region (1000 voxels -> 32 padded tile-pairs, K=432) ----
  {
    v8f cinit;
#pragma unroll
    for (int r = 0; r < 8; ++r) cinit[r] = sm.B2[8 * laneHi + r];
    for (int p = wave; p < 32; p += 8) {
      int vox0 = p * 32 + ln;
      int vox1 = vox0 + 16;
      int v0c = (vox0 < 1000) ? vox0 : 999;           // clamp pad lanes (results dropped)
      int v1c = (vox1 < 1000) ? vox1 : 999;
      int d0 = v0c / 100, r0 = v0c - d0 * 100, h0 = r0 / 10, w0 = r0 - h0 * 10;
      int d1 = v1c / 100, r1 = v1c - d1 * 100, h1 = r1 / 10, w1 = r1 - h1 * 10;
      int nB0 = (d0 * 144 + h0 * 12 + w0) * 16;
      int nB1 = (d1 * 144 + h1 * 12 + w1) * 16;
      v8f a0 = cinit, a1 = cinit;
      gemm_acc2(sm.W2, 432, sm.Off2, sm.H1, nB0, nB1, 108, laneHi, ln, a0, a1);
      if (vox0 < 1000) {
        float* dst = sm.u.H2 + vox0 * 16 + 8 * laneHi;
#pragma unroll
        for (int r = 0; r < 8; ++r) dst[r] = a0[r];
      }
      if (vox1 < 1000) {
        float* dst = sm.u.H2 + vox1 * 16 + 8 * laneHi;
#pragma unroll
        for (int r = 0; r < 8; ++r) dst[r] = a1[r];
      }
    }
  }
  __syncthreads();

  // ---- conv3: 3^3, 16->3 (M padded to 16, rows 3..15 zero), 8^3 out (16 tile-pairs) ----
  {
    v8f cinit;
#pragma unroll
    for (int r = 0; r < 8; ++r) cinit[r] = sm.B3[8 * laneHi + r];
    for (int p = wave; p < 16; p += 8) {
      int vox0 = p * 32 + ln;
      int vox1 = vox0 + 16;
      int d0 = vox0 >> 6, h0 = (vox0 >> 3) & 7, w0 = vox0 & 7;
      int d1 = vox1 >> 6, h1 = (vox1 >> 3) & 7, w1 = vox1 & 7;
      int nB0 = (d0 * 100 + h0 * 10 + w0) * 16;
      int nB1 = (d1 * 100 + h1 * 10 + w1) * 16;
      v8f a0 = cinit, a1 = cinit;
      gemm_acc2(sm.W3, 432, sm.Off3, sm.u.H2, nB0, nB1, 108, laneHi, ln, a0, a1);
      if (laneHi == 0) {                               // rows 0..7 live in lanes 0-15
        size_t base0 = (size_t)b * 3 * 884736
                     + (size_t)(Od + d0) * 9216 + (Oh + h0) * 96 + (Ow + w0);
        size_t base1 = (size_t)b * 3 * 884736
                     + (size_t)(Od + d1) * 9216 + (Oh + h1) * 96 + (Ow + w1);
#pragma unroll
        for (int r = 0; r < 3; ++r) {                  // co = r
          out[base0 + (size_t)r * 884736] = a0[r];
          out[base1 + (size_t)r * 884736] = a1[r];
        }
      }
    }
  }
}

extern "C" void kernel_launch(void* const* d_in, const int* in_sizes, int n_in,
                              void* d_out, int out_size, void* d_ws, size_t ws_size,
                              hipStream_t stream) {
  (void)in_sizes; (void)n_in; (void)d_ws; (void)ws_size; (void)out_size;
  const float* x  = (const float*)d_in[0];
  const float* W1 = (const float*)d_in[1];
  const float* b1 = (const float*)d_in[2];
  const float* W2 = (const float*)d_in[3];
  const float* b2 = (const float*)d_in[4];
  const float* W3 = (const float*)d_in[5];
  const float* b3 = (const float*)d_in[6];
  float* out = (float*)d_out;
  // 4 batches x 12^3 output tiles of 8^3, 256 threads (8 wave32) each
  fused_cnn<<<dim3(4 * 1728), dim3(NT), 0, stream>>>(x, W1, b1, W2, b2, W3, b3, out);
}